// WCCLoss_13271448944881
// MI455X (gfx1250) — compile-verified
//
#include <hip/hip_runtime.h>
#include <math.h>

// WCC loss for MI455X (gfx1250, wave32, WMMA).
//
// norm  = y_true . (h (*) y_pred)
// norm1 = y_true . (h (*) y_true)
// norm2 = y_pred . (h (*) y_pred)
// h[d] = exp(-((1-d)^2)/(2*width2)), truncated to 128 taps (error < 1e-17
// of peak, far below fp32 eps).
//
// Two adjacent 16-col tiles are produced per 160-wide slab pass:
//   G1 = A @ B(toeplitz), G2 = A @ B_shifted16, with B_shifted16[kstep kk]
//   == B[kstep kk-4] -> shared tap registers. 40 K-steps, 4 independent
//   V_WMMA_F32_16X16X4_F32 accumulation chains, B fully register-resident.

typedef __attribute__((ext_vector_type(2))) float v2f;
typedef __attribute__((ext_vector_type(8))) float v8f;

#define FDIM   2048
#define KSTEPS 36              // B rows 0..143 in steps of 4
#define PSTEPS 40              // K-steps covering the 160-wide pair slab
#define NCOLG  4               // column groups per row-group (grid.y)
#define PAIRS_PER_CG (FDIM / 32 / NCOLG)   // 16 pair-tiles (32 cols each)

#define WMMA_F32(A, B, C) \
  __builtin_amdgcn_wmma_f32_16x16x4_f32(false, (A), false, (B), (short)0, (C), false, false)

__global__ __launch_bounds__(128) void wcc_stage1(
    const float* __restrict__ yt, const float* __restrict__ yp,
    float* __restrict__ ws)
{
  // Toeplitz tap matrix in exact WMMA-B lane order:
  // Bls[(kk*32 + lane)*2 + v] = B[i = 4*kk + v + 2*(lane>>4), j = lane&15]
  __shared__ float Bls[KSTEPS * 32 * 2];
  __shared__ float red[4][3][16];

  const int tid  = threadIdx.x;
  const int lane = tid & 31;
  const int wave = tid >> 5;

  {
    double width2 = 30.0 / sqrt(2.0 * log(2.0)) * 2.0;  // HWHM=30
    float dn = (float)(2.0 * width2);                   // ~101.9186
    for (int e = tid; e < KSTEPS * 64; e += 128) {
      int kk = e >> 6;
      int v  = (e >> 5) & 1;
      int L  = e & 31;
      int i  = 4 * kk + v + 2 * (L >> 4);   // K row within the slab
      int j  = L & 15;                      // output col within tile
      int u  = 127 + j - i;                 // tap index (d = u - 63)
      float val = 0.0f;
      if (u >= 0 && u < 128) {
        float de = (float)(64 - u);         // = 1 - d
        val = expf(-(de * de) / dn);
      }
      Bls[(kk * 32 + L) * 2 + v] = val;
    }
  }
  __syncthreads();

  const int rb  = blockIdx.x * 16;          // 16-row group
  const int cg  = blockIdx.y;               // column group
  const int hk  = lane >> 4;                // half-wave (K-pair select)
  const int l15 = lane & 15;

  const float* trow = yt + (size_t)(rb + l15) * FDIM;
  const float* prow = yp + (size_t)(rb + l15) * FDIM;

  // Preload the full tap matrix into registers (36 x v2f = 72 VGPRs);
  // the hot loop then has no LDS traffic at all.
  v2f bm[KSTEPS];
  {
    const float* bptr = &Bls[lane * 2];
#pragma unroll
    for (int kk = 0; kk < KSTEPS; ++kk)
      bm[kk] = *(const v2f*)(bptr + 64 * kk);
  }

  v8f accN = {}, acc1 = {}, acc2 = {};

  for (int pi = wave; pi < PAIRS_PER_CG; pi += 4) {
    const int t0   = cg * (PAIRS_PER_CG * 32) + pi * 32;  // tile1 @ t0, tile2 @ t0+16
    const int base = t0 - 64;               // pair slab covers [base, base+159]
    v8f dt1 = {}, dp1 = {}, dt2 = {}, dp2 = {};

    if (base >= 0 && base + 160 <= FDIM) {
      // ---- interior fast path: branch-free, 8B-aligned b64 loads ----
      const float* tp = trow + base + 2 * hk;
      const float* pp = prow + base + 2 * hk;
#pragma unroll
      for (int kk = 0; kk < PSTEPS; ++kk) {
        v2f at = *(const v2f*)(tp + 4 * kk);
        v2f ap = *(const v2f*)(pp + 4 * kk);
        if (kk < KSTEPS) {                  // tile1 band (B rows >143 are 0)
          dt1 = WMMA_F32(at, bm[kk], dt1);
          dp1 = WMMA_F32(ap, bm[kk], dp1);
        }
        if (kk >= 4) {                      // tile2 band (B2 rows <16 are 0)
          dt2 = WMMA_F32(at, bm[kk - 4], dt2);
          dp2 = WMMA_F32(ap, bm[kk - 4], dp2);
        }
      }
    } else {
      // ---- edge path: per-element guards (4 edge pairs per row total) ----
#pragma unroll
      for (int kk = 0; kk < PSTEPS; ++kk) {
        int c = base + 4 * kk + 2 * hk;
        v2f at = {0.0f, 0.0f}, ap = {0.0f, 0.0f};
        if ((unsigned)c < (unsigned)FDIM)       { at.x = trow[c];     ap.x = prow[c];     }
        if ((unsigned)(c + 1) < (unsigned)FDIM) { at.y = trow[c + 1]; ap.y = prow[c + 1]; }
        if (kk < KSTEPS) {
          dt1 = WMMA_F32(at, bm[kk], dt1);
          dp1 = WMMA_F32(ap, bm[kk], dp1);
        }
        if (kk >= 4) {
          dt2 = WMMA_F32(at, bm[kk - 4], dt2);
          dp2 = WMMA_F32(ap, bm[kk - 4], dp2);
        }
      }
    }

    // D layout: d[v] lane L = smooth[rb + v + 8*(L>>4), t0(+16) + (L&15)].
    // Fuse the three dot products directly on the accumulator tiles.
    const size_t xr  = (size_t)(rb + 8 * hk) * FDIM;
    const int    xc1 = t0 + l15;
    const int    xc2 = t0 + 16 + l15;
#pragma unroll
    for (int v = 0; v < 8; ++v) {
      const size_t o = xr + (size_t)v * FDIM;
      float xt1 = yt[o + xc1], xp1 = yp[o + xc1];
      float xt2 = yt[o + xc2], xp2 = yp[o + xc2];
      accN[v] += xt1 * dp1[v];  accN[v] += xt2 * dp2[v];   // true . smooth(pred)
      acc1[v] += xt1 * dt1[v];  acc1[v] += xt2 * dt2[v];   // true . smooth(true)
      acc2[v] += xp1 * dp1[v];  acc2[v] += xp2 * dp2[v];   // pred . smooth(pred)
    }
  }

  // Reduce over the 16-lane halves: lane0 -> row rb+v, lane16 -> row rb+8+v.
#pragma unroll
  for (int v = 0; v < 8; ++v) {
    float a = accN[v], b1 = acc1[v], c2 = acc2[v];
    for (int off = 1; off < 16; off <<= 1) {
      a  += __shfl_xor(a,  off, 32);
      b1 += __shfl_xor(b1, off, 32);
      c2 += __shfl_xor(c2, off, 32);
    }
    if (lane == 0)  { red[wave][0][v]     = a; red[wave][1][v]     = b1; red[wave][2][v]     = c2; }
    if (lane == 16) { red[wave][0][8 + v] = a; red[wave][1][8 + v] = b1; red[wave][2][8 + v] = c2; }
  }
  __syncthreads();

  if (tid < 48) {
    int comp = tid >> 4, r = tid & 15;
    float s = red[0][comp][r] + red[1][comp][r] + red[2][comp][r] + red[3][comp][r];
    ws[((size_t)(rb + r) * NCOLG + cg) * 3 + comp] = s;
  }
}

__global__ __launch_bounds__(256) void wcc_stage2(
    const float* __restrict__ ws, float* __restrict__ out, int N)
{
  __shared__ float sred[256];
  const int tid = threadIdx.x;
  float s = 0.0f;
  for (int r = tid; r < N; r += 256) {
    float n = 0.0f, n1 = 0.0f, n2 = 0.0f;
    for (int cgi = 0; cgi < NCOLG; ++cgi) {
      const float* p = ws + ((size_t)r * NCOLG + cgi) * 3;
      n += p[0]; n1 += p[1]; n2 += p[2];
    }
    float sim = n / sqrtf(n1 * n2);
    sim = fminf(fmaxf(sim, 0.0f), 1.0f);
    s += sim;
  }
  sred[tid] = s;
  __syncthreads();
  for (int off = 128; off > 0; off >>= 1) {
    if (tid < off) sred[tid] += sred[tid + off];
    __syncthreads();
  }
  if (tid == 0) out[0] = 1.0f - sred[0] / (float)N;
}

extern "C" void kernel_launch(void* const* d_in, const int* in_sizes, int n_in,
                              void* d_out, int out_size, void* d_ws, size_t ws_size,
                              hipStream_t stream) {
  const float* yt = (const float*)d_in[0];   // y_true  [N, 2048] f32
  const float* yp = (const float*)d_in[1];   // y_pred  [N, 2048] f32
  float* out = (float*)d_out;                // scalar f32
  float* ws  = (float*)d_ws;                 // [N][NCOLG][3] f32 partials

  const int N = in_sizes[0] / FDIM;          // 4096
  dim3 grid(N / 16, NCOLG);
  wcc_stage1<<<grid, 128, 0, stream>>>(yt, yp, ws);
  wcc_stage2<<<1, 256, 0, stream>>>(ws, out, N);
}